// OverlapAdd1d_78116865179935
// MI455X (gfx1250) — compile-verified
//
#include <hip/hip_runtime.h>
#include <stdint.h>

// Overlap-add: y[b, f*HOP + c] += x[b, c, f]
//   x: [16, 1024, 2048] f32, y: [16, 525056] f32
// Gather formulation (each output written exactly once, no atomics):
//   y[b, p] = sum_{k=0..3} x[b, (p%256)+256k, p/256 - k]   (valid frames only)
//
// Memory-bound (~160 MiB traffic, ~33 MFLOP) -> pure data-movement kernel:
//  - CDNA5 128KB LDS tile (needs 320KB/WGP LDS)
//  - global->LDS copy done with GLOBAL_LOAD_ASYNC_TO_LDS_B32: per-lane global
//    address AND per-lane LDS address = transposed scatter with no VGPR
//    round-trip, tracked on ASYNCcnt (s_wait_asynccnt before the barrier)
//  - coalesced 128B global reads, conflict-free LDS banks, coalesced
//    non-temporal stores (write-once output kept out of L2)

#define WIN       1024
#define HOP       256
#define NFRAMES   2048
#define NBATCH    16
#define OUT_HOPS  2051              // out_len / HOP  (out_len = 2047*256+1024)
#define OUT_LEN   (OUT_HOPS * HOP)  // 525056
#define TILE_HOPS 29                // output hop-blocks produced per workgroup
#define TILE_F    32                // frames staged in LDS (TILE_HOPS + 3 overlap)
#define LDS_STR   1025              // WIN + 1 padding -> conflict-free banks
#define NTILES    71                // ceil(OUT_HOPS / TILE_HOPS)

__global__ __launch_bounds__(256)
void OverlapAdd1d_kernel(const float* __restrict__ x, float* __restrict__ y) {
    // 32 frame-columns x 1024 window-rows, stored [fcol][c], padded stride.
    __shared__ float lds[TILE_F * LDS_STR];   // 131,200 B (CDNA5-only size)

    const int t    = threadIdx.x;             // 0..255
    const int tile = blockIdx.x;              // 0..NTILES-1
    const int b    = blockIdx.y;              // 0..NBATCH-1

    const int H0 = tile * TILE_HOPS;          // first output hop-block of tile
    const int F0 = H0 - 3;                    // first frame column staged

    const float* __restrict__ xb = x + (size_t)b * WIN * NFRAMES;
    float*       __restrict__ yb = y + (size_t)b * OUT_LEN;
    const uint64_t xbase = (uint64_t)(uintptr_t)xb;   // SGPR pair for GVS mode

    // Warm L2 for the next tile's frame columns (global_prefetch_b8).
    {
        const int fNext = F0 + TILE_F;
        if (fNext < NFRAMES) {
            const int rowp = t * 4;           // rows 0..1020 step 4
            __builtin_prefetch(xb + (size_t)rowp * NFRAMES + fNext, 0, 1);
        }
    }

    // ---- Async load + transpose into LDS ----------------------------------
    // 1024 rows x 32 cols = 32768 elements; each wave reads 32 consecutive
    // floats of one row (one full 128B line) and scatters them transposed
    // into LDS directly via the async-to-LDS engine (no VGPR staging).
    #pragma unroll 4
    for (int i = 0; i < (WIN * TILE_F) / 256; ++i) {   // 128 iterations
        const int lin  = i * 256 + t;                  // 0..32767
        const int row  = lin >> 5;                     // c   : 0..1023
        const int fcol = lin & 31;                     // 0..31
        const int f    = F0 + fcol;                    // global frame index
        // banks: (fcol*1025 + row) % 64 == (fcol + row) % 64 -> conflict-free
        const uint32_t ldsAddr =
            (uint32_t)(uintptr_t)(&lds[fcol * LDS_STR + row]);
        if (f >= 0 && f < NFRAMES) {
            const uint32_t goff = (uint32_t)(((row << 11) + f) << 2); // bytes
            // copies 4 bytes from xbase plus offset goff to the LDS address in ldsAddr; per lane; ASYNCcnt
            asm volatile("global_load_async_to_lds_b32 %0, %1, %2"
                         :
                         : "v"(ldsAddr), "v"(goff), "s"(xbase)
                         : "memory");
        } else {
            lds[fcol * LDS_STR + row] = 0.0f;          // OOB frame -> zero
        }
    }

    // Drain the async copy engine, then rendezvous the workgroup.
    asm volatile("s_wait_asynccnt 0" ::: "memory");
    __syncthreads();

    // ---- Compute + store ---------------------------------------------------
    // Output p = (H0+j)*HOP + t.  Contribution k uses row c = t + 256k and
    // frame f = (H0+j) - k, i.e. LDS column (j+3-k).
    if (H0 + TILE_HOPS <= OUT_HOPS) {
        // Fast path: full tile (all tiles except the last one).
        #pragma unroll
        for (int j = 0; j < TILE_HOPS; ++j) {
            float acc = lds[(j + 3) * LDS_STR + t]               // k=0
                      + lds[(j + 2) * LDS_STR + t + 1 * HOP]     // k=1
                      + lds[(j + 1) * LDS_STR + t + 2 * HOP]     // k=2
                      + lds[(j + 0) * LDS_STR + t + 3 * HOP];    // k=3
            // write-once output: bypass near caches (th:TH_STORE_NT)
            __builtin_nontemporal_store(acc, &yb[(size_t)(H0 + j) * HOP + t]);
        }
    } else {
        // Clipped tail tile.
        for (int j = 0; j < TILE_HOPS; ++j) {
            const int h = H0 + j;
            if (h >= OUT_HOPS) break;
            float acc = lds[(j + 3) * LDS_STR + t]
                      + lds[(j + 2) * LDS_STR + t + 1 * HOP]
                      + lds[(j + 1) * LDS_STR + t + 2 * HOP]
                      + lds[(j + 0) * LDS_STR + t + 3 * HOP];
            __builtin_nontemporal_store(acc, &yb[(size_t)h * HOP + t]);
        }
    }
}

extern "C" void kernel_launch(void* const* d_in, const int* in_sizes, int n_in,
                              void* d_out, int out_size, void* d_ws, size_t ws_size,
                              hipStream_t stream) {
    const float* x = (const float*)d_in[0];
    float*       y = (float*)d_out;

    dim3 grid(NTILES, NBATCH);   // 71 x 16 = 1136 workgroups
    dim3 block(256);             // 8 wave32 per workgroup
    hipLaunchKernelGGL(OverlapAdd1d_kernel, grid, block, 0, stream, x, y);
}